// TemporalLabelAttentionClassifier_71786083385853
// MI455X (gfx1250) — compile-verified
//
#include <hip/hip_runtime.h>
#include <hip/hip_bf16.h>
#include <math.h>

// Problem constants (fixed by the reference's setup_inputs()).
#define HDIM 768
#define NCT  4096          // NC*T rows of encoding
#define NLBL 8921          // label count
#define NNOTE 4
#define TLEN 512
#define NCHUNK 32          // prefix-sum chunks (4096/128)
#define CHROWS 128
#define NTILE 48           // 768 / 16 column tiles
#define KSEG 32            // K-split segments
#define KLEN 280           // per-segment K length (32*280 = 8960 >= 8921), multiple of 4

// Workspace layout (in floats):
//   PS   : [4097][768] exclusive column prefix sums of encoding (row 4096 = total)
//   CT   : [32][768]   chunk totals -> chunk offsets
//   PART : [48][32][4][16] per-(tile,kseg) partial D tiles
//   A1T  : [8921][16]  WMMA A-matrix values w[n, j0(l)] (zero-padded rows 4..15)
//   A2T  : [8921][16]  WMMA A-matrix values w[n, j1(l)]
//   SPL  : [8921]      (int) split(l) * 768, pre-scaled PS row offset
#define PS_OFF   ((size_t)0)
#define CT_OFF   ((size_t)4097 * HDIM)
#define PART_OFF (CT_OFF + (size_t)NCHUNK * HDIM)
#define A1T_OFF  (PART_OFF + (size_t)NTILE * KSEG * NNOTE * 16)
#define A2T_OFF  (A1T_OFF + (size_t)NLBL * 16)
#define SPL_OFF  (A2T_OFF + (size_t)NLBL * 16)

typedef float v2f __attribute__((ext_vector_type(2)));
typedef float v8f __attribute__((ext_vector_type(8)));

// ---------------- Kernel A: per-chunk exclusive prefix within 128-row chunks ----
__global__ void k_prefix_local(const float* __restrict__ enc, float* __restrict__ ps,
                               float* __restrict__ ct) {
    const int c = blockIdx.x;
    const int h = blockIdx.y * 256 + threadIdx.x;
    float acc = 0.0f;
    const size_t base = (size_t)c * CHROWS * HDIM + h;
#pragma unroll 4
    for (int r = 0; r < CHROWS; ++r) {
        const size_t idx = base + (size_t)r * HDIM;
        ps[idx] = acc;          // local exclusive prefix
        acc += enc[idx];
    }
    ct[(size_t)c * HDIM + h] = acc;
}

// ---------------- Kernel B: scan the 32 chunk totals per column -----------------
__global__ void k_chunk_scan(float* __restrict__ ps, float* __restrict__ ct) {
    const int h = blockIdx.x * 256 + threadIdx.x;
    float off = 0.0f;
#pragma unroll
    for (int c = 0; c < NCHUNK; ++c) {
        const size_t idx = (size_t)c * HDIM + h;
        const float t = ct[idx];
        ct[idx] = off;          // becomes exclusive chunk offset
        off += t;
    }
    ps[(size_t)NCT * HDIM + h] = off;   // row 4096 = column total
}

// ---------------- Kernel C: add chunk offsets -> global exclusive prefix --------
__global__ void k_prefix_fix(float* __restrict__ ps, const float* __restrict__ ct) {
    const int c = blockIdx.x;
    const int h = blockIdx.y * 256 + threadIdx.x;
    const float off = ct[(size_t)c * HDIM + h];
    const size_t base = (size_t)c * CHROWS * HDIM + h;
#pragma unroll 4
    for (int r = 0; r < CHROWS; ++r)
        ps[base + (size_t)r * HDIM] += off;
}

// ---------------- Kernel F: per-l tables (note weights + split offsets) ---------
// w[n,j] = softmax-over-notes of the mask column at position j:
//   cnt = #unmasked notes; cnt==0 -> 1/4 for all n; else 1/cnt if note n unmasked, 0.
__global__ void k_build_tables(const long long* __restrict__ ids,
                               float* __restrict__ a1t, float* __restrict__ a2t,
                               int* __restrict__ spl) {
    const int l = blockIdx.x * 256 + threadIdx.x;
    if (l >= NLBL) return;

    const int b0 = ((int)ids[0] + 1) * TLEN;
    const int b1 = ((int)ids[1] + 1) * TLEN;
    const int b2 = ((int)ids[2] + 1) * TLEN;
    const int b3 = ((int)ids[3] + 1) * TLEN;
    const int bv[NNOTE] = {b0, b1, b2, b3};

    const unsigned q  = (unsigned)l << 12;        // l * 4096 (< 2^26)
    const unsigned j0 = q / 8921u;
    const unsigned rr = q - j0 * 8921u;
    unsigned split = 8921u - rr;
    if (split > 4096u) split = 4096u;
    unsigned j1 = j0 + 1u;
    if (j1 > 4095u) j1 = 4095u;                   // harmless: its B column is exactly 0 then

    spl[l] = (int)(split * HDIM);

    const int i0 = (int)j0, i1 = (int)j1;
    const int c0 = (b0 <= i0) + (b1 <= i0) + (b2 <= i0) + (b3 <= i0);
    const int c1 = (b0 <= i1) + (b1 <= i1) + (b2 <= i1) + (b3 <= i1);
    const float inv0 = 1.0f / (float)(c0 ? c0 : 1);
    const float inv1 = 1.0f / (float)(c1 ? c1 : 1);

    float* r1 = a1t + (size_t)l * 16;
    float* r2 = a2t + (size_t)l * 16;
#pragma unroll
    for (int t = 0; t < 16; ++t) {
        float w0 = 0.0f, w1 = 0.0f;
        if (t < NNOTE) {
            w0 = (c0 == 0) ? 0.25f : ((bv[t] <= i0) ? inv0 : 0.0f);
            w1 = (c1 == 0) ? 0.25f : ((bv[t] <= i1) ? inv1 : 0.0f);
        }
        r1[t] = w0;
        r2[t] = w1;
    }
}

// ---------------- Kernel D: WMMA contraction over l -----------------------------
// score[n,h] = sum_l w[n,j0(l)] * (lw[l,h]*PS[split(l),h])
//            + w[n,j1(l)] * (lw[l,h]*(tot[h]-PS[split(l),h]))
// A(16x4 f32): lane(hs,t) vgpr vi holds A[m=t, k=2*hs+vi]
// B(4x16 f32): lane(hs,t) vgpr vi holds B[k=2*hs+vi, n=t]
// D(16x16 f32): vgpr i, lanes 0-15 -> M=i
__global__ void k_wmma_main(const float* __restrict__ ps,
                            const float* __restrict__ lw,
                            const float* __restrict__ a1t,
                            const float* __restrict__ a2t,
                            const int*   __restrict__ spl,
                            float* __restrict__ part) {
    const int lane = threadIdx.x;        // wave32 block
    const int t  = lane & 15;
    const int hs = lane >> 4;
    const int h  = blockIdx.x * 16 + t;
    const int lbase = blockIdx.y * KLEN;

    const float tot = ps[(size_t)NCT * HDIM + h];

    v8f acc = {0.f, 0.f, 0.f, 0.f, 0.f, 0.f, 0.f, 0.f};

#pragma unroll 2
    for (int kk = 0; kk < KLEN; kk += 4) {
        v2f a1, a2, bv1, bv2;
#pragma unroll
        for (int vi = 0; vi < 2; ++vi) {
            const int l  = lbase + kk + 2 * hs + vi;
            const bool valid = (l < NLBL);
            const int lc = valid ? l : (NLBL - 1);   // clamp tail, mask via B=0

            const int   soff = spl[lc];              // broadcast within 16 lanes
            const float lwv  = lw[(size_t)lc * HDIM + h];
            const float psv  = ps[(size_t)soff + h];

            float m1 = lwv * psv;
            float m2 = lwv * (tot - psv);
            m1 = valid ? m1 : 0.0f;
            m2 = valid ? m2 : 0.0f;

            a1[vi]  = a1t[(size_t)lc * 16 + t];
            a2[vi]  = a2t[(size_t)lc * 16 + t];
            bv1[vi] = m1;
            bv2[vi] = m2;
        }
        // D = A*B + C, f32 16x16x4 (neg_a, A, neg_b, B, c_mod, C, reuse_a, reuse_b)
        acc = __builtin_amdgcn_wmma_f32_16x16x4_f32(false, a1, false, bv1, (short)0, acc, false, false);
        acc = __builtin_amdgcn_wmma_f32_16x16x4_f32(false, a2, false, bv2, (short)0, acc, false, false);
    }

    // D layout: vgpr i, lanes 0-15 hold row M=i. Only m<4 rows are real.
    if (hs == 0) {
        float* p = part + ((size_t)blockIdx.x * KSEG + blockIdx.y) * (NNOTE * 16);
        p[0 * 16 + t] = acc[0];
        p[1 * 16 + t] = acc[1];
        p[2 * 16 + t] = acc[2];
        p[3 * 16 + t] = acc[3];
    }
}

// ---------------- Kernel E: deterministic reduce over K segments + sigmoid ------
__global__ void k_finalize(const float* __restrict__ part, float* __restrict__ out) {
    const int idx = blockIdx.x * 256 + threadIdx.x;   // 12*256 = 3072 = 4*768
    if (idx >= NNOTE * HDIM) return;
    const int n = idx / HDIM;
    const int h = idx - n * HDIM;
    const int tile = h >> 4;
    const int col  = h & 15;
    float s = 0.0f;
#pragma unroll
    for (int k = 0; k < KSEG; ++k)
        s += part[((size_t)tile * KSEG + k) * (NNOTE * 16) + n * 16 + col];
    out[idx] = 1.0f / (1.0f + expf(-s));
}

extern "C" void kernel_launch(void* const* d_in, const int* in_sizes, int n_in,
                              void* d_out, int out_size, void* d_ws, size_t ws_size,
                              hipStream_t stream) {
    const float*     enc = (const float*)d_in[0];
    // d_in[1] (label_queries) is provably dead: softmax over the note axis cancels it exactly.
    const float*     lw  = (const float*)d_in[2];
    const long long* ids = (const long long*)d_in[3];
    float* ws  = (float*)d_ws;
    float* out = (float*)d_out;

    float* ps   = ws + PS_OFF;
    float* ct   = ws + CT_OFF;
    float* part = ws + PART_OFF;
    float* a1t  = ws + A1T_OFF;
    float* a2t  = ws + A2T_OFF;
    int*   spl  = (int*)(ws + SPL_OFF);

    k_prefix_local <<<dim3(NCHUNK, 3), 256, 0, stream>>>(enc, ps, ct);
    k_chunk_scan   <<<dim3(3),         256, 0, stream>>>(ps, ct);
    k_prefix_fix   <<<dim3(NCHUNK, 3), 256, 0, stream>>>(ps, ct);
    k_build_tables <<<dim3((NLBL + 255) / 256), 256, 0, stream>>>(ids, a1t, a2t, spl);
    k_wmma_main    <<<dim3(NTILE, KSEG), 32, 0, stream>>>(ps, lw, a1t, a2t, spl, part);
    k_finalize     <<<dim3(12),        256, 0, stream>>>(part, out);
}